// BidPrefix_69922067578885
// MI455X (gfx1250) — compile-verified
//
#include <hip/hip_runtime.h>
#include <stdint.h>

// Reference: SEQ_LEN=300, rows of 302 floats [rates(300), market_price, bid].
// Out: [survival(BATCH), rate_last(BATCH)] float32.

#define SEQ_LEN   300
#define ROW_F     302                 // floats per row
#define ROW_V2    151                 // 8-byte (v2i) packets per row (1208 B)
#define LDS_ROW_F 304                 // padded floats per wave slot (1216 B, 16B aligned)
#define WPB       8                   // waves per block
#define BLOCK_T   (WPB * 32)

typedef int v2i __attribute__((vector_size(8)));
typedef __attribute__((address_space(1))) v2i as1_v2i;
typedef __attribute__((address_space(3))) v2i as3_v2i;

__global__ __launch_bounds__(BLOCK_T)
void BidPrefix_scan_kernel(const float* __restrict__ in,
                           float* __restrict__ out,
                           int batch)
{
    __shared__ __align__(16) float smem[WPB * LDS_ROW_F];

    const int lane = threadIdx.x & 31;
    const int wv   = threadIdx.x >> 5;
    const int row  = blockIdx.x * WPB + wv;
    if (row >= batch) return;

    float*       srow = smem + wv * LDS_ROW_F;
    const float* grow = in + (size_t)row * ROW_F;

    // ---- Stage the 1208-byte row into LDS via CDNA5 async global->LDS copies.
    //      151 x 8B transfers: 4 full-wave b64 ops + one 23-lane b64 op.
    {
        as1_v2i* g = (as1_v2i*)const_cast<float*>(grow);
        as3_v2i* l = (as3_v2i*)srow;
#pragma unroll
        for (int c = 0; c < 4; ++c) {
            const int idx = c * 32 + lane;          // v2i units (8 bytes)
            __builtin_amdgcn_global_load_async_to_lds_b64(g + idx, l + idx, 0, 0);
        }
        if (lane < 23) {
            const int idx = 128 + lane;             // covers bytes [1024, 1208)
            __builtin_amdgcn_global_load_async_to_lds_b64(g + idx, l + idx, 0, 0);
        }
    }
#if __has_builtin(__builtin_amdgcn_s_wait_asynccnt)
    __builtin_amdgcn_s_wait_asynccnt(0);
#else
    asm volatile("s_wait_asynccnt 0" ::: "memory");
#endif
    asm volatile("" ::: "memory");   // compiler barrier: keep LDS reads below the wait

    // ---- Indices (same value read by all lanes; LDS broadcast).
    const int mp  = (int)srow[SEQ_LEN];       // market_price in [0, 299]
    const int bid = (int)srow[SEQ_LEN + 1];   // bid in [0, 300]
    const int k1 = bid, k2 = mp, k3 = mp + 1;

    // ---- Each lane owns 10 contiguous rates [10L, 10L+10); lanes 30/31 are padding.
    float e[10];
    if (lane < 30) {
        const float* lp = srow + lane * 10;   // 40B lane stride: bank-conflict-free
#pragma unroll
        for (int j = 0; j < 10; ++j) e[j] = lp[j];
    } else {
#pragma unroll
        for (int j = 0; j < 10; ++j) e[j] = 1.0f;
    }

    // ---- Per-lane clamped prefix products for the three targets.
    //      contrib(k) = prod of rates[i], i in [10L, min(k, 10L+10)).
    const int base = lane * 10;
    float pref = 1.0f, c1 = 1.0f, c2 = 1.0f, c3 = 1.0f;
#pragma unroll
    for (int j = 0; j < 10; ++j) {
        pref *= e[j];
        const int hi = base + j + 1;
        c1 = (k1 >= hi) ? pref : c1;
        c2 = (k2 >= hi) ? pref : c2;
        c3 = (k3 >= hi) ? pref : c3;
    }

    // ---- Wave-wide product reduction (5 xor-shuffle steps, wave32).
#pragma unroll
    for (int off = 16; off >= 1; off >>= 1) {
        c1 *= __shfl_xor(c1, off, 32);
        c2 *= __shfl_xor(c2, off, 32);
        c3 *= __shfl_xor(c3, off, 32);
    }

    if (lane == 0) {
        out[row]         = c1;        // survival_rate = cp1[bid]
        out[batch + row] = c2 - c3;   // rate_last     = cp1[mp] - cp1[mp+1]
    }
}

extern "C" void kernel_launch(void* const* d_in, const int* in_sizes, int n_in,
                              void* d_out, int out_size, void* d_ws, size_t ws_size,
                              hipStream_t stream) {
    (void)n_in; (void)out_size; (void)d_ws; (void)ws_size;
    const float* in  = (const float*)d_in[0];
    float*       out = (float*)d_out;
    const int batch  = in_sizes[0] / ROW_F;           // 500000
    const int blocks = (batch + WPB - 1) / WPB;       // 62500
    BidPrefix_scan_kernel<<<blocks, BLOCK_T, 0, stream>>>(in, out, batch);
}